// GCN_80444737454102
// MI455X (gfx1250) — compile-verified
//
#include <hip/hip_runtime.h>
#include <math.h>

// ---------------------------------------------------------------------------
// GCN (100K nodes, 1.6M edges, 4096 graphs) for MI455X / gfx1250.
//  - Dense GEMMs on V_WMMA_F32_16X16X4_F32 with a branch-free inner loop:
//    W pre-packed into per-lane fragment order (coalesced b64 loads), A read
//    as unconditional v2f from zero-padded rows, ReLU fused into the A load.
//  - Edge aggregation: TPE lanes per edge -> coalesced row gathers and
//    coalesced global_atomic_add_f32; agg buffers (<=19.2MB) are L2-resident.
// ---------------------------------------------------------------------------

constexpr int N_NODES  = 100000;
constexpr int N_EDGES  = 1600000;
constexpr int N_GRAPHS = 4096;

typedef __attribute__((ext_vector_type(2))) float v2f;
typedef __attribute__((ext_vector_type(8))) float v8f;

// --------------------------- degree precompute ------------------------------
__global__ void init_deg_kernel(float* __restrict__ deg, int n) {
    int i = blockIdx.x * blockDim.x + threadIdx.x;
    if (i < n) deg[i] = 1.0f;  // self-loop
}

__global__ void accum_deg_kernel(const int* __restrict__ dst, float* __restrict__ deg, int E) {
    int e = blockIdx.x * blockDim.x + threadIdx.x;
    if (e < E) atomicAdd(&deg[dst[e]], 1.0f);
}

__global__ void deg_inv_kernel(const float* __restrict__ deg,
                               float* __restrict__ dinv, float* __restrict__ inv, int n) {
    int i = blockIdx.x * blockDim.x + threadIdx.x;
    if (i < n) {
        float d = deg[i];
        dinv[i] = rsqrtf(d);
        inv[i]  = 1.0f / d;
    }
}

// --------------------------- W fragment packing -----------------------------
// WB[tn][s][64] : for column-tile tn, K-step s, the 64 floats are the wave's
// B-fragment in lane order: lane l holds (W[ka][col], W[ka+1][col]) where
// half = l>>4, l15 = l&15, ka = s*4 + half*2, col = tn*16 + l15.
// Zero-padded past K rows / Nw cols, so the GEMM needs no guards.
__global__ void pack_w_kernel(const float* __restrict__ W, int K, int Nw,
                              float* __restrict__ WB, int ksteps, int ntn) {
    int idx = blockIdx.x * blockDim.x + threadIdx.x;
    int total = ntn * ksteps * 64;
    if (idx >= total) return;
    int lane2 = idx & 63;
    int lane  = lane2 >> 1;
    int j     = lane2 & 1;
    int rem   = idx >> 6;
    int s     = rem % ksteps;
    int tn    = rem / ksteps;
    int half  = lane >> 4;
    int l15   = lane & 15;
    int ka    = s * 4 + half * 2 + j;
    int col   = tn * 16 + l15;
    WB[idx] = (ka < K && col < Nw) ? W[(size_t)ka * Nw + col] : 0.f;
}

// ------------------------------ WMMA GEMM -----------------------------------
// H[M x ldh] = relu?(X[M x lda]) @ W  (W pre-packed in WB fragment order).
// One wave32 per 16x16 tile; ksteps iterations of V_WMMA_F32_16X16X4_F32.
// Inner loop is branch-free: one v2f A load, one coalesced v2f B load, wmma.
// Requires: 4*ksteps <= lda with X zero-padded past K (holds for all layers),
// and M a multiple of 16 (100000 = 6250*16).
template <bool RELU_IN>
__global__ __launch_bounds__(256)
void gemm_wmma_f32(const float* __restrict__ X, int lda,
                   const float* __restrict__ WB, int ksteps,
                   float* __restrict__ H, int ldh, int ntn, int nTiles) {
    const int tile = blockIdx.x * (blockDim.x >> 5) + (threadIdx.x >> 5);
    if (tile >= nTiles) return;              // uniform per wave: EXEC all-1s

    const int tm   = tile / ntn;
    const int tn   = tile - tm * ntn;
    const int lane = threadIdx.x & 31;
    const int half = lane >> 4;
    const int l15  = lane & 15;

    const v2f* __restrict__ arow =
        (const v2f*)(X + (size_t)(tm * 16 + l15) * lda);
    const v2f* __restrict__ brow =
        (const v2f*)(WB + (size_t)tn * ksteps * 64) + lane;

    v8f c = {0.f, 0.f, 0.f, 0.f, 0.f, 0.f, 0.f, 0.f};

#pragma unroll 2
    for (int s = 0; s < ksteps; ++s) {
        v2f a = arow[s * 2 + half];
        v2f b = brow[s * 32];
        if (RELU_IN) { a.x = fmaxf(a.x, 0.f); a.y = fmaxf(a.y, 0.f); }
        // 8 args: (neg_a, A, neg_b, B, c_mod, C, reuse_a, reuse_b)
        c = __builtin_amdgcn_wmma_f32_16x16x4_f32(false, a, false, b,
                                                  (short)0, c, false, false);
    }

    // C/D layout: VGPR r -> row tm*16 + r + half*8, col tn*16 + l15.
    float* hp = H + (size_t)(tm * 16 + half * 8) * ldh + tn * 16 + l15;
#pragma unroll
    for (int r = 0; r < 8; ++r) hp[(size_t)r * ldh] = c[r];
}

// --------------------------- aggregation ------------------------------------
// agg[i,f] = h[i,f] * deg[i]^-1 + bias[f]  (self-loop + bias), padding -> 0
__global__ void agg_init_kernel(const float* __restrict__ h, const float* __restrict__ inv,
                                const float* __restrict__ bias, float* __restrict__ agg,
                                int ldh, int F, int n) {
    int idx = blockIdx.x * blockDim.x + threadIdx.x;
    int total = n * ldh;
    if (idx >= total) return;
    int i = idx / ldh;
    int f = idx - i * ldh;
    float v = 0.f;
    if (f < F) v = h[idx] * inv[i] + bias[f];
    agg[idx] = v;
}

// agg[dst,f] += h[src,f] * dinv[src]*dinv[dst].
// TPE adjacent lanes share one edge: coalesced gathers + coalesced atomics.
template <int F, int TPE>
__global__ void edge_scatter_kernel(const int* __restrict__ src, const int* __restrict__ dst,
                                    const float* __restrict__ dinv,
                                    const float* __restrict__ h, float* __restrict__ agg,
                                    int ldh, int E) {
    int idx = blockIdx.x * blockDim.x + threadIdx.x;
    int e   = idx / TPE;
    int f0  = idx & (TPE - 1);
    if (e >= E) return;
    const int s = src[e];
    const int d = dst[e];
    const float norm = dinv[s] * dinv[d];
    const float* hs = h + (size_t)s * ldh;
    float* ad = agg + (size_t)d * ldh;
#pragma unroll
    for (int f = f0; f < F; f += TPE) atomicAdd(&ad[f], hs[f] * norm);
}

// ------------------------------- pooling ------------------------------------
__global__ void pool_zero_kernel(float* __restrict__ sums, float* __restrict__ cnt) {
    int i = blockIdx.x * blockDim.x + threadIdx.x;
    if (i < N_GRAPHS * 16) sums[i] = 0.f;
    if (i < N_GRAPHS) cnt[i] = 0.f;
}

// 16 lanes per node: lane f accumulates feature f (coalesced).
__global__ void pool_scatter_kernel(const float* __restrict__ a, const int* __restrict__ batch,
                                    float* __restrict__ sums, float* __restrict__ cnt) {
    int idx = blockIdx.x * blockDim.x + threadIdx.x;
    int i = idx >> 4;
    int f = idx & 15;
    if (i >= N_NODES) return;
    int g = batch[i];
    atomicAdd(&sums[g * 16 + f], a[(size_t)i * 16 + f]);
    if (f == 0) atomicAdd(&cnt[g], 1.0f);
}

// pooled = sums/max(cnt,1); out = sigmoid((pooled@Wl1+bl1)@Wl2+bl2)
__global__ void head_kernel(const float* __restrict__ sums, const float* __restrict__ cnt,
                            const float* __restrict__ Wl1, const float* __restrict__ bl1,
                            const float* __restrict__ Wl2, const float* __restrict__ bl2,
                            float* __restrict__ out) {
    int g = blockIdx.x * blockDim.x + threadIdx.x;
    if (g >= N_GRAPHS) return;
    float ic = 1.0f / fmaxf(cnt[g], 1.0f);
    float p[16];
#pragma unroll
    for (int f = 0; f < 16; ++f) p[f] = sums[g * 16 + f] * ic;
    float t[10];
#pragma unroll
    for (int j = 0; j < 10; ++j) {
        float acc = bl1[j];
#pragma unroll
        for (int f = 0; f < 16; ++f) acc += p[f] * Wl1[f * 10 + j];
        t[j] = acc;
    }
#pragma unroll
    for (int k = 0; k < 12; ++k) {
        float acc = bl2[k];
#pragma unroll
        for (int j = 0; j < 10; ++j) acc += t[j] * Wl2[j * 12 + k];
        out[g * 12 + k] = 1.0f / (1.0f + expf(-acc));
    }
}

// ------------------------------- launcher -----------------------------------
extern "C" void kernel_launch(void* const* d_in, const int* in_sizes, int n_in,
                              void* d_out, int out_size, void* d_ws, size_t ws_size,
                              hipStream_t stream) {
    const float* x     = (const float*)d_in[0];   // [100000,100]
    const int*   edge  = (const int*)  d_in[1];   // [2,1600000]
    const int*   batch = (const int*)  d_in[2];   // [100000]
    const float* W1    = (const float*)d_in[3];   // [100,42]
    const float* b1    = (const float*)d_in[4];
    const float* W2    = (const float*)d_in[5];   // [42,24]
    const float* b2    = (const float*)d_in[6];
    const float* W3    = (const float*)d_in[7];   // [24,16]
    const float* b3    = (const float*)d_in[8];
    const float* Wl1   = (const float*)d_in[9];   // [16,10]
    const float* bl1   = (const float*)d_in[10];
    const float* Wl2   = (const float*)d_in[11];  // [10,12]
    const float* bl2   = (const float*)d_in[12];
    const int* src = edge;
    const int* dst = edge + N_EDGES;

    // workspace layout (floats): ~40 MB total
    float* ws   = (float*)d_ws;
    float* bufH = ws;                               // 100000*48  (raw xW, padded)
    float* bufA = bufH + (size_t)N_NODES * 48;      // 100000*48  (aggregated)
    float* deg  = bufA + (size_t)N_NODES * 48;      // 100000
    float* dinv = deg  + N_NODES;                   // 100000
    float* inv  = dinv + N_NODES;                   // 100000
    float* sums = inv  + N_NODES;                   // 4096*16
    float* cnt  = sums + (size_t)N_GRAPHS * 16;     // 4096
    float* WB   = cnt  + N_GRAPHS;                  // up to 3*25*64 = 4800

    const int BLK = 256;
    const int NTM = N_NODES / 16;                   // 6250 row tiles (exact)
    auto cdiv = [](long long a, long long b) { return (int)((a + b - 1) / b); };

    // degrees (deg = 1 + indegree), dinv = deg^-1/2, inv = deg^-1
    init_deg_kernel<<<cdiv(N_NODES, BLK), BLK, 0, stream>>>(deg, N_NODES);
    accum_deg_kernel<<<cdiv(N_EDGES, BLK), BLK, 0, stream>>>(dst, deg, N_EDGES);
    deg_inv_kernel<<<cdiv(N_NODES, BLK), BLK, 0, stream>>>(deg, dinv, inv, N_NODES);

    // ---- layer 1: K=100 (ksteps=25), Nw=42, ldh=48 (ntn=3) ----
    {
        const int ks = 25, ntn = 3, nt = NTM * ntn;
        pack_w_kernel<<<cdiv(ntn * ks * 64, BLK), BLK, 0, stream>>>(W1, 100, 42, WB, ks, ntn);
        gemm_wmma_f32<false><<<cdiv(nt, 8), BLK, 0, stream>>>(x, 100, WB, ks, bufH, 48, ntn, nt);
        agg_init_kernel<<<cdiv((long long)N_NODES * 48, BLK), BLK, 0, stream>>>(bufH, inv, b1, bufA, 48, 42, N_NODES);
        edge_scatter_kernel<42, 16><<<cdiv((long long)N_EDGES * 16, BLK), BLK, 0, stream>>>(src, dst, dinv, bufH, bufA, 48, N_EDGES);
    }
    // ---- layer 2: K=42 (ksteps=11, A padding zeroed by agg_init), Nw=24, ldh=32 ----
    {
        const int ks = 11, ntn = 2, nt = NTM * ntn;
        pack_w_kernel<<<cdiv(ntn * ks * 64, BLK), BLK, 0, stream>>>(W2, 42, 24, WB, ks, ntn);
        gemm_wmma_f32<true><<<cdiv(nt, 8), BLK, 0, stream>>>(bufA, 48, WB, ks, bufH, 32, ntn, nt);
        agg_init_kernel<<<cdiv((long long)N_NODES * 32, BLK), BLK, 0, stream>>>(bufH, inv, b2, bufA, 32, 24, N_NODES);
        edge_scatter_kernel<24, 8><<<cdiv((long long)N_EDGES * 8, BLK), BLK, 0, stream>>>(src, dst, dinv, bufH, bufA, 32, N_EDGES);
    }
    // ---- layer 3: K=24 (ksteps=6), Nw=16, ldh=16, no output relu ----
    {
        const int ks = 6, ntn = 1, nt = NTM;
        pack_w_kernel<<<cdiv(ntn * ks * 64, BLK), BLK, 0, stream>>>(W3, 24, 16, WB, ks, ntn);
        gemm_wmma_f32<true><<<cdiv(nt, 8), BLK, 0, stream>>>(bufA, 32, WB, ks, bufH, 16, ntn, nt);
        agg_init_kernel<<<cdiv((long long)N_NODES * 16, BLK), BLK, 0, stream>>>(bufH, inv, b3, bufA, 16, 16, N_NODES);
        edge_scatter_kernel<16, 16><<<cdiv((long long)N_EDGES * 16, BLK), BLK, 0, stream>>>(src, dst, dinv, bufH, bufA, 16, N_EDGES);
    }

    // ---- mean pool + MLP head + sigmoid ----
    pool_zero_kernel<<<cdiv(N_GRAPHS * 16, BLK), BLK, 0, stream>>>(sums, cnt);
    pool_scatter_kernel<<<cdiv((long long)N_NODES * 16, BLK), BLK, 0, stream>>>(bufA, batch, sums, cnt);
    head_kernel<<<cdiv(N_GRAPHS, BLK), BLK, 0, stream>>>(sums, cnt, Wl1, bl1, Wl2, bl2, (float*)d_out);
}